// SOMM_29429115912494
// MI455X (gfx1250) — compile-verified
//
#include <hip/hip_runtime.h>
#include <hip/hip_bf16.h>
#include <math.h>

typedef __attribute__((ext_vector_type(2))) float v2f;
typedef __attribute__((ext_vector_type(4))) float v4f;
typedef __attribute__((ext_vector_type(8))) float v8f;

#define SP_D 22
#define SP_H 11
#define CCH  1024
#define BATCH 64
#define HW   704              // 44*16
#define MROWS (BATCH * SP_H)  // 704
#define NPAD 32               // N=22 padded to two 16-wide WMMA tiles

// ---------------------------------------------------------------------------
// Kernel 0: zero-padded weight staging.  wpad[32][1024]; rows 22..31 = 0.
// Removes all per-iteration predication from the WMMA inner loop.
// ---------------------------------------------------------------------------
__global__ void pad_w_kernel(const float* __restrict__ dr_w,
                             float* __restrict__ wpad) {
    int idx = blockIdx.x * blockDim.x + threadIdx.x;   // 0 .. 32*1024-1
    int n = idx >> 10;
    wpad[idx] = (n < SP_D) ? dr_w[idx - ((n - (n < SP_D ? 0 : 0)) * 0) ] : 0.0f;
}

// (note: dr_w is [22][1024] contiguous, so for n<22 the flat index matches)

// ---------------------------------------------------------------------------
// Kernel 1: pooled-ReLU reduction, b128 streaming version.
// One wave32 per channel PAIR: 2*704 floats = 352 float4 = 11 iters * 32 lanes.
// Each 16B load sits inside one 64-float strip, so the pooled value for a
// strip is the 16-lane-half reduction of per-lane relu-hsums.  x is streamed
// exactly once with non-temporal b128 loads (184 MB -> ~8 us at 23.3 TB/s).
// ---------------------------------------------------------------------------
__global__ void pool_relu_kernel(const float* __restrict__ x,
                                 float* __restrict__ xp) {
    const int tid  = threadIdx.x;
    const int lane = tid & 31;
    const int cp   = blockIdx.x * (blockDim.x >> 5) + (tid >> 5); // 0..32767
    const v4f* base = (const v4f*)(x + (size_t)cp * (2 * HW));

#pragma unroll
    for (int i = 0; i < 11; ++i) {
        v4f v = __builtin_nontemporal_load(base + i * 32 + lane);
        float s = fmaxf(v[0], 0.0f) + fmaxf(v[1], 0.0f)
                + fmaxf(v[2], 0.0f) + fmaxf(v[3], 0.0f);
        // reduce within each 16-lane half (xor of bits 0..3 stays in-half)
        s += __shfl_xor(s, 1, 32);
        s += __shfl_xor(s, 2, 32);
        s += __shfl_xor(s, 4, 32);
        s += __shfl_xor(s, 8, 32);
        // bucket for this (iteration, half): 64-float strip
        int p0    = i * 128 + ((lane >> 4) << 6);  // start float of the half
        int chl   = p0 >= HW;                      // which channel of the pair
        int strip = (p0 - chl * HW) >> 6;
        int ch    = 2 * cp + chl;
        if ((lane & 15) == 0) {
            int b = ch >> 10, c = ch & 1023;
            xp[((size_t)(b * SP_H + strip)) * CCH + c] = s * (1.0f / 64.0f);
        }
    }
}

// ---------------------------------------------------------------------------
// Kernel 2: z = BN1( xp @ wpad^T + dr_b )  via V_WMMA_F32_16X16X4_F32.
// M=704 (44 tiles), N=32 (2 tiles, pre-padded), K=1024 (256 steps of 4).
// A 16x4 f32 fragment layout (ISA 7.12.2):
//   VGPR0: lanes 0-15 -> K=k0,   lanes 16-31 -> K=k0+2
//   VGPR1: lanes 0-15 -> K=k0+1, lanes 16-31 -> K=k0+3
// Inner loop is now pure: 2x global_load_b64 + v_wmma.
// ---------------------------------------------------------------------------
__global__ void gemm_bn_kernel(const float* __restrict__ xp,
                               const float* __restrict__ wpad,
                               const float* __restrict__ dr_b,
                               const float* __restrict__ bn1_g,
                               const float* __restrict__ bn1_b,
                               const float* __restrict__ bn1_m,
                               const float* __restrict__ bn1_v,
                               float* __restrict__ zbuf) {
    const int lane = threadIdx.x;          // blockDim.x == 32
    const int tile = blockIdx.x;           // 0..87
    const int m0 = (tile >> 1) * 16;
    const int n0 = (tile & 1) * 16;

    const int mrow  = m0 + (lane & 15);
    const int koff  = (lane >> 4) * 2;     // 0 for lanes 0-15, 2 for 16-31
    const int nn    = n0 + (lane & 15);
    const bool nval = (nn < SP_D);

    const float* arow = xp   + (size_t)mrow * CCH + koff;
    const float* brow = wpad + (size_t)nn   * CCH + koff;  // B[k][n] = wpad[n][k]

    v8f acc = {};
    for (int k = 0; k < CCH; k += 4) {
        v2f a, bfr;
        a.x   = arow[k];
        a.y   = arow[k + 1];
        bfr.x = brow[k];
        bfr.y = brow[k + 1];
        acc = __builtin_amdgcn_wmma_f32_16x16x4_f32(
                  /*neg_a=*/false, a, /*neg_b=*/false, bfr,
                  /*c_mod=*/(short)0, acc, /*reuse_a=*/false, /*reuse_b=*/false);
    }

    // fused bias + BN1 epilogue: y*s1 + ((dr_b - bn1_m)*s1 + bn1_b)
    float s1 = 0.0f, bias = 0.0f;
    if (nval) {
        s1   = bn1_g[nn] * rsqrtf(bn1_v[nn] + 1e-5f);
        bias = (dr_b[nn] - bn1_m[nn]) * s1 + bn1_b[nn];
    }
    // C/D layout: VGPR j -> M = m0 + j (+8 for lanes 16-31), N = n0 + lane&15
#pragma unroll
    for (int j = 0; j < 8; ++j) {
        int mm = m0 + j + ((lane >> 4) << 3);
        if (nval) zbuf[(size_t)mm * SP_D + nn] = acc[j] * s1 + bias;
    }
}

// ---------------------------------------------------------------------------
// Kernel 3: per-batch covariance + Newton-Schulz sqrt + row-BN + grouped conv
// + fc + sigmoid gate + broadcast to [44,16].  One 128-thread block per batch,
// 11x11 matrices ping-ponged in LDS, 121 threads = one matrix element each.
// ---------------------------------------------------------------------------
__global__ void head_kernel(const float* __restrict__ zbuf,
                            const float* __restrict__ rbn_g,
                            const float* __restrict__ rbn_b,
                            const float* __restrict__ rbn_m,
                            const float* __restrict__ rbn_v,
                            const float* __restrict__ rcw,
                            const float* __restrict__ rcb,
                            const float* __restrict__ fcw,
                            const float* __restrict__ fcb,
                            float* __restrict__ out) {
    const int b = blockIdx.x;   // 64
    const int t = threadIdx.x;  // 128

    __shared__ float z[SP_H * SP_D];        // 11 x 22
    __shared__ float rmean[SP_H];
    __shared__ float Amat[121];             // A, then An, then v (reused)
    __shared__ float M0[121], M1[121], M2[121], M3[121], M4[121], M5[121];
    __shared__ float trsh;
    __shared__ float gbuf[4 * SP_H];
    __shared__ float gate[SP_H];

    for (int i = t; i < SP_H * SP_D; i += 128)
        z[i] = zbuf[(size_t)b * SP_H * SP_D + i];
    __syncthreads();

    if (t < SP_H) {
        float s = 0.0f;
        for (int d = 0; d < SP_D; ++d) s += z[t * SP_D + d];
        rmean[t] = s * (1.0f / SP_D);
    }
    __syncthreads();
    for (int i = t; i < SP_H * SP_D; i += 128) z[i] -= rmean[i / SP_D];
    __syncthreads();

    if (t < 121) {                          // A = zc zc^T / 22
        int i = t / 11, j = t % 11;
        float s = 0.0f;
        for (int d = 0; d < SP_D; ++d) s += z[i * SP_D + d] * z[j * SP_D + d];
        Amat[t] = s * (1.0f / SP_D);
    }
    __syncthreads();
    if (t == 0) {
        float s = 0.0f;
        for (int i = 0; i < 11; ++i) s += Amat[i * 12];
        trsh = s;
    }
    __syncthreads();
    const float tr = trsh;
    if (t < 121) Amat[t] = Amat[t] / tr;    // An
    __syncthreads();

    float *Y = M0, *Z = M1, *T = M2, *W = M3, *Y2 = M4, *Z2 = M5;

    // step 0: T = 0.5*(3I - An); Y = An@T; Z = T     (diag of 11x11 at t%12==0)
    if (t < 121) {
        float tv = 0.5f * ((t % 12 == 0 ? 3.0f : 0.0f) - Amat[t]);
        T[t] = tv; Z[t] = tv;
    }
    __syncthreads();
    if (t < 121) {
        int i = t / 11, j = t % 11;
        float s = 0.0f;
        for (int k = 0; k < 11; ++k) s += Amat[i * 11 + k] * T[k * 11 + j];
        Y[t] = s;
    }
    __syncthreads();

    for (int it = 0; it < 3; ++it) {        // ITER_N - 2
        if (t < 121) {
            int i = t / 11, j = t % 11;
            float s = 0.0f;
            for (int k = 0; k < 11; ++k) s += Z[i * 11 + k] * Y[k * 11 + j];
            W[t] = s;
        }
        __syncthreads();
        if (t < 121) T[t] = 0.5f * ((t % 12 == 0 ? 3.0f : 0.0f) - W[t]);
        __syncthreads();
        if (t < 121) {
            int i = t / 11, j = t % 11;
            float s = 0.0f, s2 = 0.0f;
            for (int k = 0; k < 11; ++k) {
                s  += Y[i * 11 + k] * T[k * 11 + j];
                s2 += T[i * 11 + k] * Z[k * 11 + j];
            }
            Y2[t] = s; Z2[t] = s2;
        }
        __syncthreads();
        float* tmp;
        tmp = Y; Y = Y2; Y2 = tmp;
        tmp = Z; Z = Z2; Z2 = tmp;
    }

    // final: S = 0.5 * (Y @ (3I - Z@Y)) * sqrt(tr)  == (Y @ T) * sqrt(tr)
    if (t < 121) {
        int i = t / 11, j = t % 11;
        float s = 0.0f;
        for (int k = 0; k < 11; ++k) s += Z[i * 11 + k] * Y[k * 11 + j];
        W[t] = s;
    }
    __syncthreads();
    if (t < 121) T[t] = 0.5f * ((t % 12 == 0 ? 3.0f : 0.0f) - W[t]);
    __syncthreads();
    const float sqtr = sqrtf(tr);
    if (t < 121) {
        int i = t / 11, j = t % 11;
        float s = 0.0f;
        for (int k = 0; k < 11; ++k) s += Y[i * 11 + k] * T[k * 11 + j];
        float S  = s * sqtr;
        float rs = rbn_g[i] * rsqrtf(rbn_v[i] + 1e-5f);
        Amat[t] = (S - rbn_m[i]) * rs + rbn_b[i];    // v[c][h], row-wise BN
    }
    __syncthreads();

    if (t < 44) {                            // grouped row conv + ReLU
        int c = t >> 2, j = t & 3;
        float s = rcb[c * 4 + j];
        for (int h = 0; h < 11; ++h)
            s += Amat[c * 11 + h] * rcw[c * 44 + j * 11 + h];
        gbuf[t] = s > 0.0f ? s : 0.0f;
    }
    __syncthreads();
    if (t < SP_H) {                          // fc 44 -> 11 + sigmoid
        float s = fcb[t];
        for (int k = 0; k < 44; ++k) s += gbuf[k] * fcw[t * 44 + k];
        gate[t] = 1.0f / (1.0f + expf(-s));
    }
    __syncthreads();
    for (int i = t; i < HW; i += 128)        // broadcast: strip = (h/4) = i/64
        out[(size_t)b * HW + i] = gate[i >> 6];
}

// ---------------------------------------------------------------------------
extern "C" void kernel_launch(void* const* d_in, const int* in_sizes, int n_in,
                              void* d_out, int out_size, void* d_ws, size_t ws_size,
                              hipStream_t stream) {
    const float* x     = (const float*)d_in[0];
    const float* dr_w  = (const float*)d_in[1];
    const float* dr_b  = (const float*)d_in[2];
    const float* bn1_g = (const float*)d_in[3];
    const float* bn1_b = (const float*)d_in[4];
    const float* bn1_m = (const float*)d_in[5];
    const float* bn1_v = (const float*)d_in[6];
    const float* rbn_g = (const float*)d_in[7];
    const float* rbn_b = (const float*)d_in[8];
    const float* rbn_m = (const float*)d_in[9];
    const float* rbn_v = (const float*)d_in[10];
    const float* rcw   = (const float*)d_in[11];
    const float* rcb   = (const float*)d_in[12];
    const float* fcw   = (const float*)d_in[13];
    const float* fcb   = (const float*)d_in[14];
    float* out = (float*)d_out;

    float* xp   = (float*)d_ws;                             // 704*1024 f32
    float* zbuf = xp + (size_t)MROWS * CCH;                 // 704*22   f32
    float* wpad = zbuf + (size_t)MROWS * SP_D;              // 32*1024  f32

    // K0: stage zero-padded weights (128 KB)
    pad_w_kernel<<<(NPAD * CCH) / 256, 256, 0, stream>>>(dr_w, wpad);
    // K1: 32768 waves (one per channel pair), 8 waves/block
    pool_relu_kernel<<<(BATCH * CCH / 2) / 8, 256, 0, stream>>>(x, xp);
    // K2: 44 m-tiles x 2 n-tiles, one wave each
    gemm_bn_kernel<<<88, 32, 0, stream>>>(xp, wpad, dr_b, bn1_g, bn1_b,
                                          bn1_m, bn1_v, zbuf);
    // K3: one block per batch
    head_kernel<<<BATCH, 128, 0, stream>>>(zbuf, rbn_g, rbn_b, rbn_m, rbn_v,
                                           rcw, rcb, fcw, fcb, out);
    (void)in_sizes; (void)n_in; (void)out_size; (void)ws_size;
}